// GLIF_57724360458603
// MI455X (gfx1250) — compile-verified
//
#include <hip/hip_runtime.h>
#include <math.h>

// GLIF step, N=8192.
// Heavy op: I = I_additive @ w  (1x8192 * 8192x8192 fp32 GEMV) -> 268 MB of w,
// purely HBM-bound (~11.5 us floor @ 23.3 TB/s). We stream w through
// V_WMMA_F32_16X16X4_F32: A-matrix = w tile (documented 16x4 fp32 A layout,
// coalesced row-major reads, non-temporal since w is single-use), B-matrix =
// broadcast I_additive chunk, so each accumulator row is the partial
// dot-product of one output column.

#define NN 8192
#define SPLIT 16                          // split-K row splits
#define ROWS_PER_SPLIT (NN / SPLIT)       // 512
#define WAVES_PER_BLOCK 8
#define COLS_PER_BLOCK (WAVES_PER_BLOCK * 16)  // 128
#define UNROLL 8                          // 8 wmmas (32 rows) per pointer bump

typedef float v2f __attribute__((ext_vector_type(2)));
typedef float v8f __attribute__((ext_vector_type(8)));

// ---------------------------------------------------------------------------
// Phase 1: split-K WMMA GEMV.  partial[split][col] = sum over the split's rows
// of I_additive[i] * w[i][col].
// ---------------------------------------------------------------------------
__global__ __launch_bounds__(WAVES_PER_BLOCK * 32)
void glif_gemv_wmma(const float* __restrict__ w,
                    const float* __restrict__ I_additive,
                    float* __restrict__ partial)   // [SPLIT][NN]
{
    const int lane   = threadIdx.x & 31;
    const int wave   = threadIdx.x >> 5;
    const int j0     = blockIdx.x * COLS_PER_BLOCK + wave * 16;  // column tile
    const int col    = j0 + (lane & 15);
    const int rowOff = (lane >> 4) * 2;   // lanes 0-15 -> k{0,1}, 16-31 -> k{2,3}
    const int i0     = blockIdx.y * ROWS_PER_SPLIT;

    // A: VGPR0 = w[i+rowOff][col], VGPR1 = w[i+rowOff+1][col]
    const float* __restrict__ pa = w + (size_t)(i0 + rowOff) * NN + col;
    // B: broadcast I_additive[i+rowOff], I_additive[i+rowOff+1] (k-only value)
    const float* __restrict__ pb = I_additive + i0 + rowOff;

    v8f acc = {0.f, 0.f, 0.f, 0.f, 0.f, 0.f, 0.f, 0.f};

    for (int it = 0; it < ROWS_PER_SPLIT / (4 * UNROLL); ++it) {
#pragma unroll
        for (int u = 0; u < UNROLL; ++u) {
            v2f a, b;
            // per-chunk immediate offsets (u*4*NN*4B < 24-bit IOFFSET range)
            a.x = __builtin_nontemporal_load(pa + (size_t)(4 * u) * NN);
            a.y = __builtin_nontemporal_load(pa + (size_t)(4 * u + 1) * NN);
            b   = *(const v2f*)(pb + 4 * u);   // single b64 load
            // D = A(16x4 f32) x B(4x16 f32) + C ; every column n of D equal:
            // D[m][*] = sum_k w[i+k][j0+m] * I_additive[i+k]
            acc = __builtin_amdgcn_wmma_f32_16x16x4_f32(
                false, a, false, b, (short)0, acc, false, false);
        }
        pa += (size_t)(4 * UNROLL) * NN;
        pb += 4 * UNROLL;
    }

    // D layout: VGPR r, lanes 0-15 -> M=r, lanes 16-31 -> M=r+8 (all n equal).
    // Lane 0 emits columns j0..j0+7, lane 16 emits j0+8..j0+15.
    float* dst = partial + (size_t)blockIdx.y * NN + j0;
    if (lane == 0) {
#pragma unroll
        for (int r = 0; r < 8; ++r) dst[r] = acc[r];
    }
    if (lane == 16) {
#pragma unroll
        for (int r = 0; r < 8; ++r) dst[8 + r] = acc[r];
    }
}

// ---------------------------------------------------------------------------
// Phase 2: reduce split-K partials, remove diagonal, full GLIF neuron update.
// Outputs concatenated: spiked_soft | v_new | theta_s_new | theta_v_new | I_add_new
// ---------------------------------------------------------------------------
__global__ __launch_bounds__(256)
void glif_update(const float* __restrict__ partial,
                 const float* __restrict__ x_in,
                 const float* __restrict__ v,
                 const float* __restrict__ theta_s,
                 const float* __restrict__ theta_v,
                 const float* __restrict__ I_additive,
                 const float* __restrict__ w,
                 const float* __restrict__ E_L,
                 const float* __restrict__ tau_m,
                 const float* __restrict__ G,
                 const float* __restrict__ R_I,
                 const float* __restrict__ f_v,
                 const float* __restrict__ f_I,
                 const float* __restrict__ delta_theta_s,
                 const float* __restrict__ b_s,
                 const float* __restrict__ a_v,
                 const float* __restrict__ b_v,
                 const float* __restrict__ theta_inf,
                 const float* __restrict__ delta_V,
                 const float* __restrict__ I_A,
                 float* __restrict__ out)
{
    const int j = blockIdx.x * blockDim.x + threadIdx.x;
    if (j >= NN) return;

    float s = 0.f;
#pragma unroll
    for (int k = 0; k < SPLIT; ++k) s += partial[(size_t)k * NN + j];

    // remove diagonal contribution, add input drive
    const float Iaddj = I_additive[j];
    const float I = s - Iaddj * w[(size_t)j * NN + j] + 0.85f * x_in[j];

    const float el   = E_L[j];
    const float vj   = v[j];
    const float dv   = (I * R_I[j] - G[j] * (vj - el)) / tau_m[j];
    const float v_next = vj + dv;

    const float ths    = theta_s[j];
    const float thv    = theta_v[j];
    const float thresh = ths + thv;

    const float d = v_next - thresh;
    const float spiked_soft = 1.0f / (1.0f + __expf(-d));
    const float spiked      = (v_next >= thresh) ? 1.0f : 0.0f;
    const float not_spiked  = 1.0f - spiked;

    const float v_reset = el + f_v[j] * (vj - el) - delta_V[j];
    const float v_new   = spiked * v_reset + not_spiked * v_next;

    const float theta_s_new = (1.0f - b_s[j]) * ths + spiked * delta_theta_s[j];
    const float d_theta_v   = a_v[j] * (v_new - el) - b_v[j] * (thv - theta_inf[j]);
    const float theta_v_new = thv + not_spiked * d_theta_v;
    const float I_add_new   = (1.0f - f_I[j]) * Iaddj + spiked_soft * I_A[j];

    out[j]          = spiked_soft;
    out[NN + j]     = v_new;
    out[2 * NN + j] = theta_s_new;
    out[3 * NN + j] = theta_v_new;
    out[4 * NN + j] = I_add_new;
}

extern "C" void kernel_launch(void* const* d_in, const int* in_sizes, int n_in,
                              void* d_out, int out_size, void* d_ws, size_t ws_size,
                              hipStream_t stream) {
    const float* x_in          = (const float*)d_in[0];
    const float* v             = (const float*)d_in[1];
    const float* theta_s       = (const float*)d_in[2];
    const float* theta_v       = (const float*)d_in[3];
    const float* I_additive    = (const float*)d_in[4];
    const float* w             = (const float*)d_in[5];
    const float* E_L           = (const float*)d_in[6];
    const float* tau_m         = (const float*)d_in[7];
    const float* G             = (const float*)d_in[8];
    const float* R_I           = (const float*)d_in[9];
    const float* f_v           = (const float*)d_in[10];
    const float* f_I           = (const float*)d_in[11];
    const float* delta_theta_s = (const float*)d_in[12];
    const float* b_s           = (const float*)d_in[13];
    const float* a_v           = (const float*)d_in[14];
    const float* b_v           = (const float*)d_in[15];
    const float* theta_inf     = (const float*)d_in[16];
    const float* delta_V       = (const float*)d_in[17];
    const float* I_A           = (const float*)d_in[18];

    float* partial = (float*)d_ws;           // SPLIT * NN floats = 512 KB
    float* out     = (float*)d_out;

    dim3 g1(NN / COLS_PER_BLOCK, SPLIT);     // 64 x 16 blocks, 8 waves each
    glif_gemv_wmma<<<g1, WAVES_PER_BLOCK * 32, 0, stream>>>(w, I_additive, partial);

    glif_update<<<NN / 256, 256, 0, stream>>>(
        partial, x_in, v, theta_s, theta_v, I_additive, w, E_L, tau_m, G, R_I,
        f_v, f_I, delta_theta_s, b_s, a_v, b_v, theta_inf, delta_V, I_A, out);
}